// ChannelWiseProjection_29549374997247
// MI455X (gfx1250) — compile-verified
//
#include <hip/hip_runtime.h>

// ChannelWiseProjection as fp32 WMMA GEMM:
//   M = 4*128*128 = 65536 pixels, K = 512, N = 128 channels.
//   out[b,c,h,w] = sum_d x[b,h,w,d]*W[c,d] + bias[c]
// One wave computes a 64Mx64N macro-tile (16 accumulator tiles of 16x16)
// over the full K with V_WMMA_F32_16X16X4_F32. A workgroup (8 waves) covers
// 256M x 128N; W (256KB) is served from WGP$/L2, x streams from HBM once.

typedef __attribute__((ext_vector_type(2))) float v2f;
typedef __attribute__((ext_vector_type(4))) float v4f;
typedef __attribute__((ext_vector_type(8))) float v8f;

#define K_DIM   512
#define N_CH    128
#define HW      16384            // 128*128 pixels per batch image
#define M_TOTAL 65536            // 4 * HW
#define M_PER_WG 256

__global__ __launch_bounds__(256)
void ChannelWiseProjection_wmma_f32(const float* __restrict__ X,     // [M, K]
                                    const float* __restrict__ W,     // [N, K]
                                    const float* __restrict__ Bias,  // [N]
                                    float* __restrict__ Out)         // [4, N, HW]
{
    const int lane = threadIdx.x & 31;
    const int wave = threadIdx.x >> 5;
    const int lo   = lane & 15;   // row/col within 16-wide tile
    const int hi   = lane >> 4;   // 0 -> K pair {k,k+1}, 1 -> {k+2,k+3}

    const int wave_m = wave >> 1;                              // 0..3
    const int wave_n = wave & 1;                               // 0..1
    const int m_base = blockIdx.x * M_PER_WG + wave_m * 64;    // 64 M rows / wave
    const int n_base = wave_n * 64;                            // 64 N cols / wave

    // Per-lane row base pointers for A (x rows) and B (W rows, since B = W^T).
    // 32-bit A 16x4 layout: lane holds a contiguous float2 at K offset 2*hi.
    const float* xrow[4];
    const float* wrow[4];
#pragma unroll
    for (int t = 0; t < 4; ++t) {
        xrow[t] = X + (size_t)(m_base + t * 16 + lo) * K_DIM + 2 * hi;
        wrow[t] = W + (size_t)(n_base + t * 16 + lo) * K_DIM + 2 * hi;
    }

    v8f acc[4][4];
#pragma unroll
    for (int mt = 0; mt < 4; ++mt)
#pragma unroll
        for (int nt = 0; nt < 4; ++nt)
            acc[mt][nt] = (v8f){0.f, 0.f, 0.f, 0.f, 0.f, 0.f, 0.f, 0.f};

#pragma unroll 2
    for (int k = 0; k < K_DIM; k += 4) {
        v2f a[4], bb[4];
#pragma unroll
        for (int mt = 0; mt < 4; ++mt)
            a[mt] = *(const v2f*)(xrow[mt] + k);
#pragma unroll
        for (int nt = 0; nt < 4; ++nt)
            bb[nt] = *(const v2f*)(wrow[nt] + k);

#pragma unroll
        for (int mt = 0; mt < 4; ++mt)
#pragma unroll
            for (int nt = 0; nt < 4; ++nt)
                acc[mt][nt] = __builtin_amdgcn_wmma_f32_16x16x4_f32(
                    /*neg_a=*/false, a[mt],
                    /*neg_b=*/false, bb[nt],
                    /*c_mod=*/(short)0, acc[mt][nt],
                    /*reuse_a=*/false, /*reuse_b=*/false);
    }

    // Epilogue: add bias, store with the b,h,w,c -> b,c,h,w transpose.
    // C/D layout: VGPR r of acc holds (m_local = r + 8*hi, n_local = lo),
    // so each lane's 8 values are 8 consecutive hw positions of one channel.
#pragma unroll
    for (int nt = 0; nt < 4; ++nt) {
        const int chan = n_base + nt * 16 + lo;
        const float bv = Bias[chan];
#pragma unroll
        for (int mt = 0; mt < 4; ++mt) {
            v8f r = acc[mt][nt];
#pragma unroll
            for (int i = 0; i < 8; ++i) r[i] += bv;

            const int m_tile = m_base + mt * 16;       // multiple of 16; HW%16==0
            const int batch  = m_tile / HW;            // tile never crosses batch
            const int within = (m_tile - batch * HW) + 8 * hi;
            float* dst = Out + (size_t)batch * N_CH * HW
                             + (size_t)chan * HW + within;
            *(v4f*)(dst)     = (v4f){r[0], r[1], r[2], r[3]};
            *(v4f*)(dst + 4) = (v4f){r[4], r[5], r[6], r[7]};
        }
    }
}

extern "C" void kernel_launch(void* const* d_in, const int* in_sizes, int n_in,
                              void* d_out, int out_size, void* d_ws, size_t ws_size,
                              hipStream_t stream) {
    const float* x    = (const float*)d_in[0];  // [4,128,128,512]
    const float* W    = (const float*)d_in[1];  // [128,512]
    const float* bias = (const float*)d_in[2];  // [128]
    float* out        = (float*)d_out;          // [4,128,128,128] as b,c,h,w

    dim3 grid(M_TOTAL / M_PER_WG);   // 256 workgroups
    dim3 block(256);                 // 8 waves of 32
    ChannelWiseProjection_wmma_f32<<<grid, block, 0, stream>>>(x, W, bias, out);
}